// YoloLoss_67388036874752
// MI455X (gfx1250) — compile-verified
//
#include <hip/hip_runtime.h>
#include <hip/hip_bf16.h>

typedef __attribute__((ext_vector_type(2))) float v2f;
typedef __attribute__((ext_vector_type(8))) float v8f;

#define SGRID 14
#define INV_S (1.0f / 14.0f)

// Reduce 256 per-thread partials (in sdata) to one value using the f32 matrix
// pipe: D = ones(16x4) x B(4x16) + C contracts 64 f32 values into 16 column
// sums per WMMA. Layout details of B don't matter: with A==1 everywhere,
// every one of the 64 B slots is summed into exactly one of 16 columns, and
// columns are a bijection onto lanes 0..15 of each D VGPR. 4 accumulating
// WMMAs cover all 256 partials; 4 xor-shuffles finish the last 16.
// Must be called with all 256 threads converged (uses __syncthreads).
__device__ __forceinline__ float block_reduce_wmma(float partial, float* sdata) {
    const int tid = threadIdx.x;
    sdata[tid] = partial;
    __syncthreads();
    float total = 0.0f;
    if (tid < 32) {                 // wave 0, EXEC all ones within the wave
        const int lane = tid;
        v2f ones_a;
        ones_a[0] = 1.0f;
        ones_a[1] = 1.0f;
        v8f acc = {};
#pragma unroll
        for (int off = 0; off < 256; off += 64) {
            v2f b;
            b[0] = sdata[off + lane];
            b[1] = sdata[off + lane + 32];
            acc = __builtin_amdgcn_wmma_f32_16x16x4_f32(
                false, ones_a, false, b, (short)0, acc, false, false);
        }
        // acc[0] at lane L holds column sum c_{L%16} (rows are constant per
        // column). Sum lanes 0..15 into lane 0 with 4 xor-shuffles (masks<16
        // keep the butterfly inside the 16-lane group).
        float v = acc[0];
        v += __shfl_xor(v, 1, 32);
        v += __shfl_xor(v, 2, 32);
        v += __shfl_xor(v, 4, 32);
        v += __shfl_xor(v, 8, 32);
        total = v;
    }
    return total;                   // valid in thread 0 only
}

// One thread per (b,i,j) cell: 30 pred + 30 target floats, 8B-aligned
// float2 loads (cell stride 120 B is 8B aligned), all math in registers.
__global__ void yolo_cell_kernel(const float* __restrict__ pred,
                                 const float* __restrict__ targ,
                                 float* __restrict__ block_out,
                                 int ncells) {
    __shared__ float sdata[256];
    const int cell = blockIdx.x * blockDim.x + threadIdx.x;

    float cell_loss = 0.0f;
    if (cell < ncells) {
        float p[30], t[30];
        const float2* pv = (const float2*)(pred + (size_t)cell * 30);
        const float2* tv = (const float2*)(targ + (size_t)cell * 30);
#pragma unroll
        for (int i = 0; i < 15; ++i) {
            float2 a = pv[i];
            p[2 * i] = a.x;
            p[2 * i + 1] = a.y;
        }
#pragma unroll
        for (int i = 0; i < 15; ++i) {
            float2 a = tv[i];
            t[2 * i] = a.x;
            t[2 * i + 1] = a.y;
        }

        const float obj   = (t[4] > 0.0f) ? 1.0f : 0.0f;
        const float noobj = (t[4] == 0.0f) ? 1.0f : 0.0f;

        // no-object confidence loss (both boxes' conf slots)
        const float d4 = p[4] - t[4];
        const float d9 = p[9] - t[9];
        const float no_object = noobj * (d4 * d4 + d9 * d9);

        // class loss over channels 10..29
        float cls = 0.0f;
#pragma unroll
        for (int c = 10; c < 30; ++c) {
            const float d = p[c] - t[c];
            cls += d * d;
        }

        // IoU of each pred box vs target box 0
        const float tx = t[0] * INV_S, ty = t[1] * INV_S;
        const float tw = t[2], th = t[3];
        const float t1x = tx - 0.5f * tw, t1y = ty - 0.5f * th;
        const float t2x = tx + 0.5f * tw, t2y = ty + 0.5f * th;
        const float area_t = tw * th;

        float iou[2];
#pragma unroll
        for (int b = 0; b < 2; ++b) {
            const float px = p[5 * b + 0] * INV_S, py = p[5 * b + 1] * INV_S;
            const float pw = p[5 * b + 2], ph = p[5 * b + 3];
            const float p1x = px - 0.5f * pw, p1y = py - 0.5f * ph;
            const float p2x = px + 0.5f * pw, p2y = py + 0.5f * ph;
            const float ltx = fmaxf(p1x, t1x), lty = fmaxf(p1y, t1y);
            const float rbx = fminf(p2x, t2x), rby = fminf(p2y, t2y);
            const float wx = fmaxf(rbx - ltx, 0.0f);
            const float wy = fmaxf(rby - lty, 0.0f);
            const float inter = wx * wy;
            iou[b] = inter / (pw * ph + area_t - inter);
        }

        const bool pick1 = (iou[0] <= iou[1]);
        float psel[5], tsel[5];
#pragma unroll
        for (int i = 0; i < 5; ++i) {
            psel[i] = pick1 ? p[5 + i] : p[i];
            tsel[i] = pick1 ? t[5 + i] : t[i];
        }
        const float iou_best = pick1 ? iou[1] : iou[0];

        const float dx = psel[0] - tsel[0];
        const float dy = psel[1] - tsel[1];
        const float dw = sqrtf(psel[2]) - sqrtf(tsel[2]);
        const float dh = sqrtf(psel[3]) - sqrtf(tsel[3]);
        const float reg = dx * dx + dy * dy + dw * dw + dh * dh;

        const float dconf = psel[4] - iou_best;
        const float contain = dconf * dconf;

        cell_loss = 0.5f * no_object + obj * (cls + 5.0f * reg + contain);
    }

    const float total = block_reduce_wmma(cell_loss, sdata);
    if (threadIdx.x == 0) block_out[blockIdx.x] = total;
}

// Single block: fold all per-block partials, WMMA-reduce, scale by 1/N.
__global__ void yolo_reduce_kernel(const float* __restrict__ block_out,
                                   float* __restrict__ out,
                                   int nblocks, float inv_n) {
    __shared__ float sdata[256];
    float partial = 0.0f;
    for (int i = threadIdx.x; i < nblocks; i += 256) partial += block_out[i];
    const float total = block_reduce_wmma(partial, sdata);
    if (threadIdx.x == 0) out[0] = total * inv_n;
}

extern "C" void kernel_launch(void* const* d_in, const int* in_sizes, int n_in,
                              void* d_out, int out_size, void* d_ws, size_t ws_size,
                              hipStream_t stream) {
    const float* pred = (const float*)d_in[0];
    const float* targ = (const float*)d_in[1];
    float* out = (float*)d_out;
    float* ws = (float*)d_ws;

    const long long total_elems = (long long)in_sizes[0];   // N*S*S*30
    const int ncells = (int)(total_elems / 30);             // N*S*S
    const int nblocks = (ncells + 255) / 256;
    const int batch_n = ncells / (SGRID * SGRID);

    yolo_cell_kernel<<<nblocks, 256, 0, stream>>>(pred, targ, ws, ncells);
    yolo_reduce_kernel<<<1, 256, 0, stream>>>(ws, out, nblocks,
                                              1.0f / (float)batch_n);
}